// KernelConv_27711128994618
// MI455X (gfx1250) — compile-verified
//
#include <hip/hip_runtime.h>

// KPN dynamic per-pixel 15x15 filtering, fp32, MI455X (gfx1250). v3
//
// Roofline: kernels tensor = 199 MB read exactly once -> HBM-bound,
// ~8.5 us floor @ 23.3 TB/s. data = 0.88 MB -> 2.8 KB zero-padded LDS
// tile per block (keeps occupancy high; Little's law needs many waves +
// counter-free prefetch run-ahead to cover HBM latency).
// Contraction: V_WMMA_F32_16X16X4_F32, 16 consecutive-w pixels on both
// M and N, diagonal of D = the 16 outputs.
// Hot loop is FULLY UNROLLED: tap offsets come from a 225-entry LDS
// table via ds_load_b64 with immediate offsets; kernel stream uses
// non-temporal b32 loads + unconditional global_prefetch_b8 lookahead
// (prefetch doesn't count against LOADcnt -> free MLP).

typedef __attribute__((ext_vector_type(2))) float v2f;
typedef __attribute__((ext_vector_type(8))) float v8f;

#define TT 8
#define CC 3
#define HH 96
#define WW 96
#define KSZ 15
#define K2 225            // 56 chunks of K=4 cover taps 0..223; tap 224 scalar
#define ROWS_PER_BLK 8    // 8 waves -> 8 rows
#define GROUPS_PER_ROW 6  // 96 / 16
#define TILE_H 22         // ROWS_PER_BLK + 14 halo
#define TILE_W 32         // 16 + 14 halo, padded to 32
#define KSTRIDE (CC * HH * WW)   // 27648 elements between taps

__global__ __launch_bounds__(256)
void KernelConv_kpn_wmma(const float* __restrict__ data,
                         const float* __restrict__ kernels,
                         float* __restrict__ out)
{
    __shared__ float tile[TILE_H * TILE_W];  // 2816 B
    __shared__ int   offs[228];              // byte offset of tap k in tile (+pad)

    const int lane = threadIdx.x & 31;
    const int wv   = threadIdx.x >> 5;       // wave id == row offset in band

    // block -> (t, c, h0, w0)
    int b = blockIdx.x;
    const int w0 = (b % GROUPS_PER_ROW) * 16;
    int tmp = b / GROUPS_PER_ROW;
    const int h0 = (tmp % (HH / ROWS_PER_BLK)) * ROWS_PER_BLK;
    tmp /= (HH / ROWS_PER_BLK);
    const int c = tmp % CC;
    const int t = tmp / CC;

    const float* dplane = data + (long)(t * CC + c) * HH * WW;

    // ---- stage zero-padded 22x32 halo tile into LDS ----
    for (int i = threadIdx.x; i < TILE_H * TILE_W; i += 256) {
        const int r  = i >> 5;
        const int cc = i & 31;
        const int y  = h0 - 7 + r;
        const int x  = w0 - 7 + cc;
        float v = 0.0f;
        if (cc < 30 && y >= 0 && y < HH && x >= 0 && x < WW)
            v = dplane[y * WW + x];
        tile[i] = v;
    }
    // ---- tap -> tile byte-offset table (one-time tiny cost) ----
    if (threadIdx.x < 228) {
        int k  = threadIdx.x < K2 ? threadIdx.x : (K2 - 1);
        int di = k / KSZ, dj = k - di * KSZ;
        offs[threadIdx.x] = (di * TILE_W + dj) * 4;
    }
    __syncthreads();

    // ---- per-wave WMMA sweep ----
    const int m    = lane & 15;   // pixel inside group (M and N role)
    const int half = lane >> 4;   // 0: taps k,k+1   1: taps k+2,k+3
    const int h    = h0 + wv;
    const int w    = w0 + m;

    // per-lane byte base into the tile for (row wv, col m)
    const char* tbase = (const char*)tile + (wv * TILE_W + m) * 4;
    // per-lane offset-table base: taps (4q + 2*half), (4q + 2*half + 1)
    const int* obase = offs + 2 * half;

    // per-lane kernel stream pointer at tap (2*half)
    const float* kbase = kernels
        + ((long)t * K2 + 2 * half) * KSTRIDE
        + (long)c * HH * WW + (long)h * WW + w;

    v8f acc = {};
    #pragma unroll
    for (int q = 0; q < 56; ++q) {
        // two tap offsets, consecutive ints -> ds_load_b64, imm offset 16q
        const int o0 = obase[4 * q];
        const int o1 = obase[4 * q + 1];

        // A: kernels, pure 199MB stream -> non-temporal b32 pair (one clause)
        v2f A;
        A[0] = __builtin_nontemporal_load(kbase + (long)(4 * q) * KSTRIDE);
        A[1] = __builtin_nontemporal_load(kbase + (long)(4 * q + 1) * KSTRIDE);

        // free run-ahead: global_prefetch_b8 (no LOADcnt), pulls into GL2
        if (q < 48)
            __builtin_prefetch(kbase + (long)(4 * q + 32) * KSTRIDE, 0, 0);

        // B: branch-free LDS gather from the padded tile
        v2f B;
        B[0] = *(const float*)(tbase + o0);
        B[1] = *(const float*)(tbase + o1);

        // fp32 WMMA: diagonal of D accumulates the 16 dot products
        acc = __builtin_amdgcn_wmma_f32_16x16x4_f32(
                  false, A, false, B, (short)0, acc, false, false);
    }

    // ---- extract diagonal D[p][p] ----
    // lanes 0-15: VGPR r = (M=r, N=lane); lanes 16-31: (M=r+8, N=lane-16)
    float o = 0.0f;
    #pragma unroll
    for (int r = 0; r < 8; ++r) {
        float dv = acc[r];
        float g1 = __shfl(dv, r, 32);       // D[r][r]
        float g2 = __shfl(dv, r + 24, 32);  // D[r+8][r+8]
        if (lane == r)     o = g1;
        if (lane == r + 8) o = g2;
    }

    // ---- tap 224 (di=14, dj=14) scalar tail, then store (lanes 0..15) ----
    if (lane < 16) {
        const float* k224 = kernels
            + ((long)t * K2 + 224) * KSTRIDE
            + (long)c * HH * WW + (long)h * WW + w;
        float kv = __builtin_nontemporal_load(k224);
        float tv = tile[(wv + 14) * TILE_W + m + 14];
        o += kv * tv;
        out[((long)(t * CC + c) * HH + h) * WW + w] = o;
    }
}

extern "C" void kernel_launch(void* const* d_in, const int* in_sizes, int n_in,
                              void* d_out, int out_size, void* d_ws, size_t ws_size,
                              hipStream_t stream) {
    (void)in_sizes; (void)n_in; (void)d_ws; (void)ws_size; (void)out_size;
    const float* data    = (const float*)d_in[0];
    const float* kernels = (const float*)d_in[1];
    float* out = (float*)d_out;

    // blocks = T*C*(H/8)*(W/16) = 8*3*12*6 = 1728, 256 threads (8 waves) each
    const int blocks = TT * CC * (HH / ROWS_PER_BLK) * GROUPS_PER_ROW;
    KernelConv_kpn_wmma<<<blocks, 256, 0, stream>>>(data, kernels, out);
}